// ElasticityAttention_20538533609913
// MI455X (gfx1250) — compile-verified
//
#include <hip/hip_runtime.h>
#include <float.h>

// Problem constants (from reference)
#define N_NODES   50000
#define N_EDGES   800000
#define DIM       256
#define HEADS     8
#define CPH       32
#define EAUG      (N_EDGES + N_NODES)   // 850000 augmented edges (with self loops)
#define NEG_SLOPE 0.2f

typedef float v2f __attribute__((ext_vector_type(2)));
typedef float v8f __attribute__((ext_vector_type(8)));

// ---------------- small helpers ----------------

__device__ __forceinline__ void atomicMaxFloat(float* addr, float value) {
    // sign-aware int-punned float max (init value is -FLT_MAX)
    if (value >= 0.0f)
        atomicMax((int*)addr, __float_as_int(value));
    else
        atomicMin((unsigned int*)addr, __float_as_uint(value));
}

__global__ void k_fill(float* __restrict__ p, float v, int n) {
    int i = blockIdx.x * 256 + threadIdx.x;
    if (i < n) p[i] = v;
}

__global__ void k_init_out(float* __restrict__ out, const float* __restrict__ bias) {
    int i = blockIdx.x * 256 + threadIdx.x;
    if (i < N_NODES * DIM) out[i] = bias[i & (DIM - 1)];
}

// ---------------- self-loop attr: scatter mean ----------------

__global__ void k_edge_mean(const int* __restrict__ ei, const float* __restrict__ ea,
                            float* __restrict__ attr_sum, float* __restrict__ cnt) {
    int e = blockIdx.x * 256 + threadIdx.x;
    if (e >= N_EDGES) return;
    int d = ei[N_EDGES + e];                 // edge_index[1][e]
    atomicAdd(&attr_sum[d * 2 + 0], ea[e * 2 + 0]);
    atomicAdd(&attr_sum[d * 2 + 1], ea[e * 2 + 1]);
    atomicAdd(&cnt[d], 1.0f);
}

__global__ void k_mean_div(float* __restrict__ attr_sum, const float* __restrict__ cnt) {
    int i = blockIdx.x * 256 + threadIdx.x;
    if (i >= N_NODES) return;
    float c = fmaxf(cnt[i], 1.0f);
    attr_sum[i * 2 + 0] /= c;                // in place: attr_sum becomes loop_attr
    attr_sum[i * 2 + 1] /= c;
}

// ---------------- fp32 WMMA GEMM: x_l = x@W_l, x_r = x@W_r ----------------
// One block per 16-row tile of x (50000 = 3125 * 16, exact).
// 8 waves; each wave computes four 16x16 output tiles (32 tiles = 16 N-tiles x 2 matrices).
// K = 256 -> 64 x V_WMMA_F32_16X16X4_F32 per tile.

__global__ void __launch_bounds__(256) k_gemm_wmma(const float* __restrict__ x,
                                                   const float* __restrict__ Wl,
                                                   const float* __restrict__ Wr,
                                                   float* __restrict__ xl,
                                                   float* __restrict__ xr) {
    __shared__ float sA[16][DIM + 4];        // +4 pad: bank-conflict-free column reads
    const int m0  = blockIdx.x * 16;
    const int tid = threadIdx.x;

    // cooperative, coalesced load of the 16x256 A tile
    for (int i = tid; i < 16 * DIM; i += 256) {
        int r = i >> 8, c = i & (DIM - 1);
        sA[r][c] = x[(m0 + r) * DIM + c];
    }
    __syncthreads();

    const int lane = tid & 31;
    const int wave = tid >> 5;
    const int r    = lane & 15;              // A: row M / B: col N (within tile)
    const int koff = (lane >> 4) << 1;       // lanes 16-31 hold K+2,K+3

    for (int t = wave; t < 32; t += 8) {     // wave-uniform branch -> EXEC all-1s for WMMA
        const float* __restrict__ W   = (t < 16) ? Wl : Wr;
        float* __restrict__       Out = (t < 16) ? xl : xr;
        const int nbase = (t & 15) << 4;

        v8f c = {};
        #pragma unroll 4
        for (int kb = 0; kb < DIM; kb += 4) {
            v2f a, b;
            a[0] = sA[r][kb + koff];
            a[1] = sA[r][kb + koff + 1];
            b[0] = W[(kb + koff)     * DIM + nbase + r];
            b[1] = W[(kb + koff + 1) * DIM + nbase + r];
            // (neg_a, A, neg_b, B, c_mod, C, reuse_a, reuse_b)
            c = __builtin_amdgcn_wmma_f32_16x16x4_f32(false, a, false, b,
                                                      (short)0, c, false, false);
        }
        // C/D layout: VGPR i -> M = i (lanes 0-15) / M = i+8 (lanes 16-31), N = lane&15
        const int rhalf = (lane >> 4) << 3;
        #pragma unroll
        for (int i = 0; i < 8; ++i)
            Out[(m0 + rhalf + i) * DIM + nbase + r] = c[i];
    }
}

// ---------------- GATv2 score: one wave per (edge, head) ----------------

__global__ void __launch_bounds__(256) k_score(const int* __restrict__ ei,
                                               const float* __restrict__ edge_attr,
                                               const float* __restrict__ loop_attr,
                                               const float* __restrict__ xl,
                                               const float* __restrict__ xr,
                                               const float* __restrict__ We,
                                               const float* __restrict__ att,
                                               float* __restrict__ score,
                                               float* __restrict__ smax) {
    const int p = blockIdx.x * 8 + (threadIdx.x >> 5);   // (edge, head) pair index
    if (p >= EAUG * HEADS) return;
    const int e    = p >> 3;
    const int h    = p & 7;
    const int lane = threadIdx.x & 31;                   // channel c

    int s, d; float ea0, ea1;                            // wave-uniform branch
    if (e < N_EDGES) {
        s = ei[e]; d = ei[N_EDGES + e];
        ea0 = edge_attr[e * 2]; ea1 = edge_attr[e * 2 + 1];
    } else {
        s = e - N_EDGES; d = s;
        ea0 = loop_attr[s * 2]; ea1 = loop_attr[s * 2 + 1];
    }

    const int hc = h * CPH + lane;
    float m = xl[s * DIM + hc] + xr[d * DIM + hc]
            + ea0 * We[hc] + ea1 * We[DIM + hc];
    m = (m > 0.0f) ? m : NEG_SLOPE * m;                  // leaky_relu
    float v = m * att[hc];

    #pragma unroll
    for (int o = 16; o > 0; o >>= 1) v += __shfl_xor(v, o, 32);

    if (lane == 0) {
        score[p] = v;
        atomicMaxFloat(&smax[d * HEADS + h], v);
    }
}

// ---------------- segment softmax: exp + denom ----------------

__global__ void k_exp(const int* __restrict__ ei, float* __restrict__ score,
                      const float* __restrict__ smax, float* __restrict__ denom) {
    int p = blockIdx.x * 256 + threadIdx.x;
    if (p >= EAUG * HEADS) return;
    int e = p >> 3, h = p & 7;
    int d = (e < N_EDGES) ? ei[N_EDGES + e] : (e - N_EDGES);
    float ex = __expf(score[p] - smax[d * HEADS + h]);
    score[p] = ex;                                       // overwrite score with exp
    atomicAdd(&denom[d * HEADS + h], ex);
}

// ---------------- weighted scatter-add: one block per edge ----------------

__global__ void __launch_bounds__(256) k_scatter(const int* __restrict__ ei,
                                                 const float* __restrict__ xl,
                                                 const float* __restrict__ ex,
                                                 const float* __restrict__ denom,
                                                 float* __restrict__ out) {
    int e  = blockIdx.x;
    int hc = threadIdx.x;
    int h  = hc >> 5;
    int s, d;
    if (e < N_EDGES) { s = ei[e]; d = ei[N_EDGES + e]; }
    else             { s = e - N_EDGES; d = s; }
    float alpha = ex[e * HEADS + h] / denom[d * HEADS + h];
    atomicAdd(&out[d * DIM + hc], alpha * xl[s * DIM + hc]);
}

// ---------------- launch ----------------

extern "C" void kernel_launch(void* const* d_in, const int* in_sizes, int n_in,
                              void* d_out, int out_size, void* d_ws, size_t ws_size,
                              hipStream_t stream) {
    const float* x         = (const float*)d_in[0];
    const int*   ei        = (const int*)  d_in[1];
    const float* edge_attr = (const float*)d_in[2];
    const float* Wl        = (const float*)d_in[3];
    const float* Wr        = (const float*)d_in[4];
    const float* We        = (const float*)d_in[5];
    const float* att       = (const float*)d_in[6];
    const float* bias      = (const float*)d_in[7];
    float*       out       = (float*)d_out;

    // Workspace layout (floats): ~134 MB total
    float* ws       = (float*)d_ws;
    float* attr_sum = ws;                                    // N*2  (becomes loop_attr)
    float* cnt      = attr_sum + 2 * N_NODES;                // N
    float* smax     = cnt + N_NODES;                         // N*H
    float* denom    = smax + (size_t)N_NODES * HEADS;        // N*H
    float* xl       = denom + (size_t)N_NODES * HEADS;       // N*DIM
    float* xr       = xl + (size_t)N_NODES * DIM;            // N*DIM
    float* score    = xr + (size_t)N_NODES * DIM;            // EAUG*H

    hipMemsetAsync(attr_sum, 0, sizeof(float) * 3 * N_NODES, stream);          // attr_sum + cnt
    hipMemsetAsync(denom, 0, sizeof(float) * (size_t)N_NODES * HEADS, stream);

    k_fill<<<(N_NODES * HEADS + 255) / 256, 256, 0, stream>>>(smax, -FLT_MAX, N_NODES * HEADS);
    k_init_out<<<(N_NODES * DIM + 255) / 256, 256, 0, stream>>>(out, bias);

    k_edge_mean<<<(N_EDGES + 255) / 256, 256, 0, stream>>>(ei, edge_attr, attr_sum, cnt);
    k_mean_div<<<(N_NODES + 255) / 256, 256, 0, stream>>>(attr_sum, cnt);

    k_gemm_wmma<<<N_NODES / 16, 256, 0, stream>>>(x, Wl, Wr, xl, xr);

    k_score<<<(EAUG * HEADS) / 8, 256, 0, stream>>>(ei, edge_attr, attr_sum,
                                                    xl, xr, We, att, score, smax);
    k_exp<<<(EAUG * HEADS + 255) / 256, 256, 0, stream>>>(ei, score, smax, denom);
    k_scatter<<<EAUG, 256, 0, stream>>>(ei, xl, score, denom, out);
}